// GINPhi_44573170597936
// MI455X (gfx1250) — compile-verified
//
#include <hip/hip_runtime.h>
#include <stddef.h>

// ---------------------------------------------------------------------------
// GIN forward for MI455X (gfx1250):
//   - scatter-add aggregation via global f32 atomics (block-per-edge, scalar idx)
//   - per-node 2-layer MLP via v_wmma_f32_16x16x4_f32 (exact f32 math)
//   - X tiles staged with global_load_async_to_lds_b128 (ASYNCcnt path)
//   - layer-2 fuses the 64-row sum-pool
// ---------------------------------------------------------------------------

typedef float v2f __attribute__((ext_vector_type(2)));
typedef float v8f __attribute__((ext_vector_type(8)));

#define N_SUM  6400
#define N_MAX  64
#define MDIM   16
#define HDIM   32
#define ODIM   32
#define EDGES  51200

// agg = (1 + eps[layer]) * h   (elementwise, float4)
__global__ void scale_kernel(const float* __restrict__ in, float* __restrict__ out,
                             const float* __restrict__ eps, int layer, long n4) {
    long i = (long)blockIdx.x * blockDim.x + threadIdx.x;
    if (i >= n4) return;
    float s = 1.0f + eps[layer];
    float4 v = ((const float4*)in)[i];
    v.x *= s; v.y *= s; v.z *= s; v.w *= s;
    ((float4*)out)[i] = v;
}

// agg[dst] += h[src] over 64*D floats; one block per edge so src/dst are
// uniform scalar loads and the inner loop is pure b128-load + atomic_add_f32.
template<int D>
__global__ __launch_bounds__(256) void scatter_kernel(
        const float* __restrict__ h, float* __restrict__ agg,
        const int* __restrict__ src, const int* __restrict__ dst) {
    const int e = blockIdx.x;
    const int s = src[e];                       // uniform -> s_load
    const int d = dst[e];
    const float* hp = h  + (long)s * (N_MAX * D);
    float*       ap = agg + (long)d * (N_MAX * D);
    #pragma unroll
    for (int o = threadIdx.x * 4; o < N_MAX * D; o += 256 * 4) {
        float4 v = *(const float4*)(hp + o);
        atomicAdd(ap + o + 0, v.x);
        atomicAdd(ap + o + 1, v.y);
        atomicAdd(ap + o + 2, v.z);
        atomicAdd(ap + o + 3, v.w);
    }
}

// Per-node 2-layer MLP: Y = [relu]( relu(X@W1+b1) @ W2 + b2 ), optional pool.
// Block = 128 threads (4 waves); wave w owns rows [w*16, w*16+16) of the node.
// WMMA f32 16x16x4 layouts (ISA 7.12.2):
//   A 16x4 : lane L -> M = L&15, VGPR{0,1} = K = 2*(L>>4) + {0,1}
//   B 4x16 : lane L -> N = L&15, VGPR{0,1} = K = 2*(L>>4) + {0,1}
//   C 16x16: lane L -> N = L&15, VGPR r   = M = r + 8*(L>>4)
template<int KIN, bool RELU_OUT, bool POOL>
__global__ __launch_bounds__(128) void mlp_kernel(
        const float* __restrict__ X,   // [N_SUM*64, KIN]
        const float* __restrict__ W1,  // [KIN, 32] row-major
        const float* __restrict__ b1,  // [32]
        const float* __restrict__ W2,  // [32, 32]
        const float* __restrict__ b2,  // [32]
        float* __restrict__ Y)         // POOL ? [N_SUM,32] : [N_SUM*64,32]
{
    constexpr int XS = KIN + 4;        // pad: 16B-aligned rows, conflict-free frags
    constexpr int HS = HDIM + 2;
    __shared__ __align__(16) float xt[N_MAX * XS];
    __shared__ __align__(16) float ht[4 * 16 * HS];
    __shared__ float pool[ODIM];

    const int tid  = threadIdx.x;
    const int wave = tid >> 5;
    const int lane = tid & 31;
    const int node = blockIdx.x;
    const long xbase = (long)node * (N_MAX * KIN);

    // --- stage node's 64xKIN tile into LDS via CDNA5 async DMA (ASYNCcnt) ---
    {
        unsigned long long gbase = (unsigned long long)(X + xbase);
        #pragma unroll
        for (int i = tid * 4; i < N_MAX * KIN; i += 128 * 4) {
            int r = i / KIN, c = i % KIN;          // KIN % 4 == 0: no row crossing
            unsigned ldsoff = (unsigned)(size_t)(&xt[r * XS + c]);
            unsigned goff   = (unsigned)(i * 4);   // byte offset from gbase
            asm volatile("global_load_async_to_lds_b128 %0, %1, %2"
                         :: "v"(ldsoff), "v"(goff), "s"(gbase) : "memory");
        }
        asm volatile("s_wait_asynccnt 0x0" ::: "memory");
    }
    if (POOL && tid < ODIM) pool[tid] = 0.0f;
    __syncthreads();

    const int lrow  = lane & 15;               // M (A/C) or N (B/C) index
    const int khalf = (lane >> 4) * 2;         // K sub-offset within a frag

    // ---- GEMM1: hidden = relu(Xtile @ W1 + b1), two 16-wide N tiles ----
    v8f acc0, acc1;
    {
        float bb0 = b1[lrow], bb1 = b1[16 + lrow];
        #pragma unroll
        for (int r = 0; r < 8; ++r) { acc0[r] = bb0; acc1[r] = bb1; }
    }
    const float* xrow = xt + (wave * 16 + lrow) * XS;
    #pragma unroll
    for (int k0 = 0; k0 < KIN; k0 += 4) {
        v2f a;  a.x = xrow[k0 + khalf];  a.y = xrow[k0 + khalf + 1];
        v2f bA, bB;
        bA.x = W1[(k0 + khalf) * HDIM + lrow];
        bA.y = W1[(k0 + khalf + 1) * HDIM + lrow];
        bB.x = W1[(k0 + khalf) * HDIM + 16 + lrow];
        bB.y = W1[(k0 + khalf + 1) * HDIM + 16 + lrow];
        acc0 = __builtin_amdgcn_wmma_f32_16x16x4_f32(false, a, false, bA, (short)0, acc0, false, false);
        acc1 = __builtin_amdgcn_wmma_f32_16x16x4_f32(false, a, false, bB, (short)0, acc1, false, false);
    }

    // ReLU, write hidden 16x32 tile to this wave's private LDS region
    float* hw = ht + wave * 16 * HS;
    #pragma unroll
    for (int r = 0; r < 8; ++r) {
        int m = r + 8 * (lane >> 4);
        float h0 = acc0[r] > 0.0f ? (float)acc0[r] : 0.0f;
        float h1 = acc1[r] > 0.0f ? (float)acc1[r] : 0.0f;
        hw[m * HS + lrow]      = h0;
        hw[m * HS + 16 + lrow] = h1;
    }
    // per-wave LDS region: in-wave DS ordering suffices, no __syncthreads needed

    // ---- GEMM2: out = hidden @ W2 + b2 ----
    v8f o0, o1;
    {
        float bb0 = b2[lrow], bb1 = b2[16 + lrow];
        #pragma unroll
        for (int r = 0; r < 8; ++r) { o0[r] = bb0; o1[r] = bb1; }
    }
    const float* hrow = hw + lrow * HS;
    #pragma unroll
    for (int k0 = 0; k0 < HDIM; k0 += 4) {
        v2f a;  a.x = hrow[k0 + khalf];  a.y = hrow[k0 + khalf + 1];
        v2f bA, bB;
        bA.x = W2[(k0 + khalf) * ODIM + lrow];
        bA.y = W2[(k0 + khalf + 1) * ODIM + lrow];
        bB.x = W2[(k0 + khalf) * ODIM + 16 + lrow];
        bB.y = W2[(k0 + khalf + 1) * ODIM + 16 + lrow];
        o0 = __builtin_amdgcn_wmma_f32_16x16x4_f32(false, a, false, bA, (short)0, o0, false, false);
        o1 = __builtin_amdgcn_wmma_f32_16x16x4_f32(false, a, false, bB, (short)0, o1, false, false);
    }

    if (POOL) {
        // fused sum-pool over all 64 rows of the node (mask is all-true)
        float s0 = 0.0f, s1 = 0.0f;
        #pragma unroll
        for (int r = 0; r < 8; ++r) { s0 += o0[r]; s1 += o1[r]; }
        atomicAdd(&pool[lrow], s0);
        atomicAdd(&pool[16 + lrow], s1);
        __syncthreads();
        if (tid < ODIM) Y[(long)node * ODIM + tid] = pool[tid];
    } else {
        long rowbase = (long)node * N_MAX + wave * 16;
        #pragma unroll
        for (int r = 0; r < 8; ++r) {
            int m = r + 8 * (lane >> 4);
            float v0 = o0[r], v1 = o1[r];
            if (RELU_OUT) { v0 = v0 > 0.0f ? v0 : 0.0f; v1 = v1 > 0.0f ? v1 : 0.0f; }
            Y[(rowbase + m) * ODIM + lrow]      = v0;
            Y[(rowbase + m) * ODIM + 16 + lrow] = v1;
        }
    }
}

extern "C" void kernel_launch(void* const* d_in, const int* in_sizes, int n_in,
                              void* d_out, int out_size, void* d_ws, size_t ws_size,
                              hipStream_t stream) {
    (void)in_sizes; (void)n_in; (void)out_size; (void)ws_size;

    const float* W    = (const float*)d_in[0];
    const int*   ei   = (const int*)d_in[1];
    const float* eps  = (const float*)d_in[2];
    const float* W1_0 = (const float*)d_in[3];
    const float* b1_0 = (const float*)d_in[4];
    const float* W2_0 = (const float*)d_in[5];
    const float* b2_0 = (const float*)d_in[6];
    const float* W1_1 = (const float*)d_in[7];
    const float* b1_1 = (const float*)d_in[8];
    const float* W2_1 = (const float*)d_in[9];
    const float* b2_1 = (const float*)d_in[10];
    const float* W1_2 = (const float*)d_in[11];
    const float* b1_2 = (const float*)d_in[12];
    const float* W2_2 = (const float*)d_in[13];
    const float* b2_2 = (const float*)d_in[14];

    const int* src = ei;
    const int* dst = ei + EDGES;

    const long FWIDE = (long)N_SUM * N_MAX * HDIM;   // 13,107,200 floats
    float* bufA = (float*)d_ws;
    float* bufB = bufA + FWIDE;
    float* bufC = bufB + FWIDE;

    // ---------------- Layer 0 (D=16) ----------------
    {
        long n4 = (long)N_SUM * N_MAX * MDIM / 4;
        scale_kernel<<<(unsigned)((n4 + 255) / 256), 256, 0, stream>>>(W, bufA, eps, 0, n4);
        scatter_kernel<MDIM><<<EDGES, 256, 0, stream>>>(W, bufA, src, dst);
        mlp_kernel<MDIM, true, false><<<N_SUM, 128, 0, stream>>>(bufA, W1_0, b1_0, W2_0, b2_0, bufB);
    }
    // ---------------- Layer 1 (D=32) ----------------
    {
        long n4 = (long)N_SUM * N_MAX * HDIM / 4;
        scale_kernel<<<(unsigned)((n4 + 255) / 256), 256, 0, stream>>>(bufB, bufC, eps, 1, n4);
        scatter_kernel<HDIM><<<EDGES, 256, 0, stream>>>(bufB, bufC, src, dst);
        mlp_kernel<HDIM, true, false><<<N_SUM, 128, 0, stream>>>(bufC, W1_1, b1_1, W2_1, b2_1, bufA);
    }
    // ---------------- Layer 2 (D=32) + fused pool ----------------
    {
        long n4 = (long)N_SUM * N_MAX * HDIM / 4;
        scale_kernel<<<(unsigned)((n4 + 255) / 256), 256, 0, stream>>>(bufA, bufB, eps, 2, n4);
        scatter_kernel<HDIM><<<EDGES, 256, 0, stream>>>(bufA, bufB, src, dst);
        mlp_kernel<HDIM, false, true><<<N_SUM, 128, 0, stream>>>(bufB, W1_2, b1_2, W2_2, b2_2, (float*)d_out);
    }
}